// GroupClassifier_46832323396450
// MI455X (gfx1250) — compile-verified
//
#include <hip/hip_runtime.h>
#include <hip/hip_bf16.h>
#include <math.h>

// ---------------------------------------------------------------------------
// GroupClassifier (graph transformer) for MI455X / gfx1250, wave32.
// GEMMs: V_WMMA_F32_16X16X4_F32 with double-buffered LDS staging of W via
// async global->LDS loads (ASYNCcnt) when available.
// Segment softmax / scatter use global float atomics (L2-native).
// ---------------------------------------------------------------------------

typedef float v2f __attribute__((ext_vector_type(2)));
typedef float v8f __attribute__((ext_vector_type(8)));
typedef int   vi4 __attribute__((vector_size(16)));   // int4 vector for async-LDS builtin

#define HID 256
#define NGRAPH 64
#define LN_EPS 1e-5f
#define INV_SQRT_HD 0.17677669529663687f  // 1/sqrt(32)

#define GLOBAL_AS __attribute__((address_space(1)))
#define LDS_AS    __attribute__((address_space(3)))

#if __has_builtin(__builtin_amdgcn_global_load_async_to_lds_b128)
#define HAVE_ASYNC_LDS 1
#endif

// ---- monotonic float<->uint encoding for atomicMax on floats ----
__device__ __forceinline__ unsigned encF(float f) {
    unsigned u = __float_as_uint(f);
    return (u & 0x80000000u) ? ~u : (u | 0x80000000u);
}
__device__ __forceinline__ float decF(unsigned u) {
    u = (u & 0x80000000u) ? (u & 0x7fffffffu) : ~u;
    return __uint_as_float(u);
}

__device__ __forceinline__ float waveAllSum(float v) {
    #pragma unroll
    for (int m = 1; m < 32; m <<= 1) v += __shfl_xor(v, m, 32);
    return v;
}

// async (or sync fallback) 16B global -> LDS stage
__device__ __forceinline__ void stageW(const float* g, float* l) {
#ifdef HAVE_ASYNC_LDS
    __builtin_amdgcn_global_load_async_to_lds_b128(
        (GLOBAL_AS vi4*)g, (LDS_AS vi4*)l, 0, 0);
#else
    float4 t = *(const float4*)g;
    *(float4*)l = t;
#endif
}
__device__ __forceinline__ void waitAsync() {
#ifdef HAVE_ASYNC_LDS
#if __has_builtin(__builtin_amdgcn_s_wait_asynccnt)
    __builtin_amdgcn_s_wait_asynccnt(0);
#else
    asm volatile("s_wait_asynccnt 0x0" ::: "memory");
#endif
#endif
}

// ---------------------------------------------------------------------------
// C[M,Nc] = A[M,K] @ W[K,Nc] + bias  (opt relu / accumulate)
// Block = 8 waves, tile 128(M) x 64(N). Wave w: rows [bm*128+w*16, +16),
// 4 accumulators over the 64-col strip. W staged 16x64 per chunk into
// double-buffered LDS (async). WMMA fragments per ISA f32 layouts:
//   A 16x4 : lane&15 = M, VGPR j = A[M][2*(lane>=16)+j]
//   B 4x16 : lane&15 = N, VGPR j = B[2*(lane>=16)+j][N]
//   C 16x16: VGPR r -> row r + 8*(lane>=16), col = lane&15
// Requires: K % 16 == 0, Nc % 64 == 0. M arbitrary (clamped/masked).
// ---------------------------------------------------------------------------
#define GEMM_RELU 1
#define GEMM_ACC  2
#define BM 128
#define BN 64
#define BK 16

__global__ __launch_bounds__(256) void gemm_lds(
    const float* __restrict__ A, const float* __restrict__ W,
    const float* __restrict__ bias, float* __restrict__ C,
    int M, int K, int Nc, int flags)
{
    __shared__ float Wbuf[2][BK * BN];   // 2 x 4KB

    const int tid  = threadIdx.x;
    const int lane = tid & 31;
    const int wv   = tid >> 5;           // wave id 0..7 -> M band
    const int lr   = lane & 15;
    const int hi   = lane >> 4;

    const int nBlocksN = Nc / BN;
    const int bm = (int)blockIdx.x / nBlocksN;
    const int bn = (int)blockIdx.x - bm * nBlocksN;
    const int row0 = bm * BM + wv * 16;
    const int col0 = bn * BN;

    // A fragment row (clamped so loads stay in-bounds when M < band end)
    int arow = row0 + lr;
    if (arow > M - 1) arow = M - 1;
    const float* Arow = A + (size_t)arow * K;

    // cooperative W staging: 16 rows x 64 cols, one float4 per thread
    const int wr = tid >> 4;             // 0..15 chunk row
    const int wc = (tid & 15) * 4;       // 0..60 col
    const float* Wg = W + (size_t)wr * Nc + (col0 + wc);
    const int ldsoff = wr * BN + wc;

    v8f acc[4];
    if (flags & GEMM_ACC) {
        #pragma unroll
        for (int nt = 0; nt < 4; ++nt) {
            #pragma unroll
            for (int r = 0; r < 8; ++r) {
                int rr = row0 + r + 8 * hi;
                acc[nt][r] = (rr < M) ? C[(size_t)rr * Nc + col0 + nt * 16 + lr] : 0.f;
            }
        }
    } else {
        #pragma unroll
        for (int nt = 0; nt < 4; ++nt)
            acc[nt] = (v8f){0.f, 0.f, 0.f, 0.f, 0.f, 0.f, 0.f, 0.f};
    }

    const int nChunks = K / BK;
    stageW(Wg, &Wbuf[0][ldsoff]);        // chunk 0 -> buf 0
    waitAsync();
    __syncthreads();

    for (int c2 = 0; c2 < nChunks; ++c2) {
        const int buf = c2 & 1;
        if (c2 + 1 < nChunks)
            stageW(Wg + (size_t)(c2 + 1) * BK * Nc, &Wbuf[buf ^ 1][ldsoff]);

        const float* Ak = Arow + c2 * BK;
        #pragma unroll
        for (int kk = 0; kk < BK; kk += 4) {
            int ka = kk + 2 * hi;
            float2 av = *(const float2*)(Ak + ka);
            v2f a; a.x = av.x; a.y = av.y;
            #pragma unroll
            for (int nt = 0; nt < 4; ++nt) {
                v2f b;
                b.x = Wbuf[buf][ka * BN + nt * 16 + lr];
                b.y = Wbuf[buf][(ka + 1) * BN + nt * 16 + lr];
                acc[nt] = __builtin_amdgcn_wmma_f32_16x16x4_f32(
                              false, a, false, b, (short)0, acc[nt], false, false);
            }
        }
        waitAsync();
        __syncthreads();
    }

    #pragma unroll
    for (int nt = 0; nt < 4; ++nt) {
        float bb = bias[col0 + nt * 16 + lr];
        #pragma unroll
        for (int r = 0; r < 8; ++r) {
            int rr = row0 + r + 8 * hi;
            if (rr < M) {
                float v = acc[nt][r] + bb;
                if (flags & GEMM_RELU) v = fmaxf(v, 0.f);
                C[(size_t)rr * Nc + col0 + nt * 16 + lr] = v;
            }
        }
    }
}

// ---------------------------------------------------------------------------
__global__ void fill_u32(unsigned* p, unsigned v, long n) {
    long i = (long)blockIdx.x * blockDim.x + threadIdx.x;
    if (i < n) p[i] = v;
}

// embed: h[N,256] = x[N,5] @ W + b   (K=5: not worth WMMA)
__global__ void embed_kernel(const float* __restrict__ x, const float* __restrict__ W,
                             const float* __restrict__ b, float* __restrict__ h, int n) {
    int idx = blockIdx.x * blockDim.x + threadIdx.x;
    if (idx >= n * HID) return;
    int node = idx >> 8, c = idx & 255;
    const float* xr = x + (size_t)node * 5;
    float acc = b[c];
    #pragma unroll
    for (int j = 0; j < 5; ++j) acc += xr[j] * W[j * HID + c];
    h[idx] = acc;
}

// layernorm over 256 channels; one wave per row
__global__ void layernorm_kernel(const float* __restrict__ in, float* __restrict__ out,
                                 const float* __restrict__ g, const float* __restrict__ b, int n) {
    int gtid = blockIdx.x * blockDim.x + threadIdx.x;
    int row = gtid >> 5, lane = gtid & 31;
    if (row >= n) return;
    const float* rp = in + (size_t)row * HID;
    float vals[8], s = 0.f;
    #pragma unroll
    for (int j = 0; j < 8; ++j) { vals[j] = rp[j * 32 + lane]; s += vals[j]; }
    float mu = waveAllSum(s) * (1.f / 256.f);
    float s2 = 0.f;
    #pragma unroll
    for (int j = 0; j < 8; ++j) { float d = vals[j] - mu; s2 += d * d; }
    float var = waveAllSum(s2) * (1.f / 256.f);
    float rs = rsqrtf(var + LN_EPS);
    float* op = out + (size_t)row * HID;
    #pragma unroll
    for (int j = 0; j < 8; ++j) {
        int c = j * 32 + lane;
        op[c] = (vals[j] - mu) * rs * g[c] + b[c];
    }
}

// attention pass 1: scores + running max per (dst, head); one wave per edge.
// channel mapping c = j*32+lane -> head == j.
__global__ void attn_score(const float* __restrict__ q, const float* __restrict__ k,
                           const float* __restrict__ ea, const float* __restrict__ We,
                           const float* __restrict__ be,
                           const int* __restrict__ src, const int* __restrict__ dst,
                           float* __restrict__ alpha, unsigned* __restrict__ mEnc, int E_) {
    int gtid = blockIdx.x * blockDim.x + threadIdx.x;
    int e = gtid >> 5, lane = gtid & 31;
    if (e >= E_) return;
    int sN = src[e], dN = dst[e];
    const float* qd = q + (size_t)dN * HID;
    const float* ks = k + (size_t)sN * HID;
    float a0 = ea[(size_t)e * 4 + 0], a1 = ea[(size_t)e * 4 + 1];
    float a2 = ea[(size_t)e * 4 + 2], a3 = ea[(size_t)e * 4 + 3];
    float myScore = 0.f;
    #pragma unroll
    for (int j = 0; j < 8; ++j) {
        int c = j * 32 + lane;
        float ev = be[c] + a0 * We[c] + a1 * We[256 + c] + a2 * We[512 + c] + a3 * We[768 + c];
        float p = qd[c] * (ks[c] + ev);
        float hs = waveAllSum(p);           // head-j score (all lanes)
        if (lane == j) myScore = hs * INV_SQRT_HD;
    }
    if (lane < 8) {
        alpha[(size_t)e * 8 + lane] = myScore;
        atomicMax(&mEnc[(size_t)dN * 8 + lane], encF(myScore));
    }
}

// attention pass 2: exp(score - max) and per-(dst,head) sum
__global__ void attn_expsum(const int* __restrict__ dst, float* __restrict__ alpha,
                            const unsigned* __restrict__ mEnc, float* __restrict__ ssum, int E_) {
    int idx = blockIdx.x * blockDim.x + threadIdx.x;
    if (idx >= E_ * 8) return;
    int e = idx >> 3, h = idx & 7;
    int d = dst[e];
    float w = __expf(alpha[idx] - decF(mEnc[(size_t)d * 8 + h]));
    alpha[idx] = w;
    atomicAdd(&ssum[(size_t)d * 8 + h], w);
}

// attention pass 3: out[dst] += coef * (v[src] + e); one wave per edge
__global__ void attn_aggregate(const float* __restrict__ v, const float* __restrict__ ea,
                               const float* __restrict__ We, const float* __restrict__ be,
                               const int* __restrict__ src, const int* __restrict__ dst,
                               const float* __restrict__ alpha, const float* __restrict__ ssum,
                               float* __restrict__ out, int E_) {
    int gtid = blockIdx.x * blockDim.x + threadIdx.x;
    int e = gtid >> 5, lane = gtid & 31;
    if (e >= E_) return;
    int sN = src[e], dN = dst[e];
    const float* vs = v + (size_t)sN * HID;
    float* od = out + (size_t)dN * HID;
    float a0 = ea[(size_t)e * 4 + 0], a1 = ea[(size_t)e * 4 + 1];
    float a2 = ea[(size_t)e * 4 + 2], a3 = ea[(size_t)e * 4 + 3];
    #pragma unroll
    for (int j = 0; j < 8; ++j) {
        int c = j * 32 + lane;
        float coef = alpha[(size_t)e * 8 + j] / (ssum[(size_t)dN * 8 + j] + 1e-16f);
        float ev = be[c] + a0 * We[c] + a1 * We[256 + c] + a2 * We[512 + c] + a3 * We[768 + c];
        atomicAdd(&od[c], coef * (vs[c] + ev));
    }
}

// beta gate: h += beta*xr + (1-beta)*attn ; one wave per node
__global__ void beta_combine(const float* __restrict__ attn, const float* __restrict__ xr,
                             const float* __restrict__ Wb, float* __restrict__ h, int n) {
    int gtid = blockIdx.x * blockDim.x + threadIdx.x;
    int node = gtid >> 5, lane = gtid & 31;
    if (node >= n) return;
    float o[8], r[8], p = 0.f;
    #pragma unroll
    for (int j = 0; j < 8; ++j) {
        int c = j * 32 + lane;
        o[j] = attn[(size_t)node * HID + c];
        r[j] = xr[(size_t)node * HID + c];
        p += o[j] * Wb[c] + r[j] * Wb[256 + c] + (o[j] - r[j]) * Wb[512 + c];
    }
    p = waveAllSum(p);
    float beta = 1.f / (1.f + __expf(-p));
    #pragma unroll
    for (int j = 0; j < 8; ++j) {
        int c = j * 32 + lane;
        h[(size_t)node * HID + c] += beta * r[j] + (1.f - beta) * o[j];
    }
}

__global__ void copy_cat(const float* __restrict__ h, float* __restrict__ xcat, int n, int blk) {
    int idx = blockIdx.x * blockDim.x + threadIdx.x;
    if (idx >= n * HID) return;
    int node = idx >> 8, c = idx & 255;
    xcat[(size_t)node * 512 + blk * HID + c] = h[idx];
}

// gate score = gh[n] . Wg2 + b ; track per-graph max
__global__ void gate_score(const float* __restrict__ gh, const float* __restrict__ W,
                           const float* __restrict__ b, const int* __restrict__ batch,
                           float* __restrict__ score, unsigned* __restrict__ gmax, int n) {
    int gtid = blockIdx.x * blockDim.x + threadIdx.x;
    int node = gtid >> 5, lane = gtid & 31;
    if (node >= n) return;
    float p = 0.f;
    #pragma unroll
    for (int j = 0; j < 8; ++j) {
        int c = j * 32 + lane;
        p += gh[(size_t)node * HID + c] * W[c];
    }
    p = waveAllSum(p) + b[0];
    if (lane == 0) {
        score[node] = p;
        atomicMax(&gmax[batch[node]], encF(p));
    }
}

__global__ void gate_expsum(const int* __restrict__ batch, float* __restrict__ score,
                            const unsigned* __restrict__ gmax, float* __restrict__ gsum, int n) {
    int idx = blockIdx.x * blockDim.x + threadIdx.x;
    if (idx >= n) return;
    int g = batch[idx];
    float w = __expf(score[idx] - decF(gmax[g]));
    score[idx] = w;
    atomicAdd(&gsum[g], w);
}

__global__ void pool_scatter(const float* __restrict__ xcat, const float* __restrict__ w,
                             const float* __restrict__ gsum, const int* __restrict__ batch,
                             float* __restrict__ pooled, int n) {
    int gtid = blockIdx.x * blockDim.x + threadIdx.x;
    int node = gtid >> 5, lane = gtid & 31;
    if (node >= n) return;
    int g = batch[node];
    float coef = w[node] / (gsum[g] + 1e-16f);
    #pragma unroll
    for (int j = 0; j < 16; ++j) {
        int c = j * 32 + lane;
        atomicAdd(&pooled[(size_t)g * 512 + c], coef * xcat[(size_t)node * 512 + c]);
    }
}

// head2: out[64,3] = headh[64,256] @ W[256,3] + b
__global__ void head2_kernel(const float* __restrict__ hh, const float* __restrict__ W,
                             const float* __restrict__ b, float* __restrict__ out) {
    int idx = threadIdx.x;
    if (idx >= NGRAPH * 3) return;
    int g = idx / 3, c = idx % 3;
    float acc = b[c];
    for (int k2 = 0; k2 < 256; ++k2) acc += hh[(size_t)g * 256 + k2] * W[k2 * 3 + c];
    out[idx] = acc;
}

// ---------------------------------------------------------------------------
extern "C" void kernel_launch(void* const* d_in, const int* in_sizes, int n_in,
                              void* d_out, int out_size, void* d_ws, size_t ws_size,
                              hipStream_t stream) {
    const float* x     = (const float*)d_in[0];
    const int*   eidx  = (const int*)d_in[1];
    const float* eattr = (const float*)d_in[2];
    const int*   batch = (const int*)d_in[3];
    const int N_ = in_sizes[0] / 5;
    const int E_ = in_sizes[1] / 2;
    const int* src = eidx;
    const int* dst = eidx + E_;

    // jax tree_leaves order: dict keys sorted. params -> blocks, embed, gate1,
    // gate2, head1, head2. Block keys: Wbeta, e, f1, f2, k, ln1_b, ln1_g,
    // ln2_b, ln2_g, q, skip, v; each linear {W, b} -> W first. 19 leaves/block.
    auto P = [&](int i) { return (const float*)d_in[i]; };
    const int B0 = 4, BSTRIDE = 19;
    const int I_EMBED = 4 + 2 * BSTRIDE;   // 42

    // ---- workspace layout (floats) ----
    float* ws = (float*)d_ws;
    size_t o = 0;
    float* H   = ws + o; o += (size_t)N_ * 256;
    float* HN  = ws + o; o += (size_t)N_ * 256;
    float* Qb  = ws + o; o += (size_t)N_ * 256;   // FFH aliases Qb..XR (N*1024)
    float* Kb  = ws + o; o += (size_t)N_ * 256;
    float* Vb  = ws + o; o += (size_t)N_ * 256;
    float* XR  = ws + o; o += (size_t)N_ * 256;
    float* ATT = ws + o; o += (size_t)N_ * 256;   // also gate-hidden later
    float* ALPHA = ws + o; o += (size_t)E_ * 8;
    unsigned* MB = (unsigned*)(ws + o); o += (size_t)N_ * 8;
    float* SB  = ws + o; o += (size_t)N_ * 8;
    float* XCAT = ws + o; o += (size_t)N_ * 512;
    float* GSC = ws + o; o += (size_t)N_;
    unsigned* GMAX = (unsigned*)(ws + o); o += NGRAPH;
    float* GSUM = ws + o; o += NGRAPH;
    float* POOL = ws + o; o += (size_t)NGRAPH * 512;
    float* HH  = ws + o; o += (size_t)NGRAPH * 256;
    float* FFH = Qb;

    const int TPB = 256;
    auto gElem = [&](long n) { return (unsigned)((n + TPB - 1) / TPB); };
    auto gWave = [&](long waves) { return (unsigned)((waves + 7) / 8); };
    auto gGemm = [&](int M, int Nc) { return (unsigned)(((M + BM - 1) / BM) * (Nc / BN)); };

    embed_kernel<<<gElem((long)N_ * 256), TPB, 0, stream>>>(x, P(I_EMBED), P(I_EMBED + 1), H, N_);

    for (int b = 0; b < 2; ++b) {
        int base = B0 + b * BSTRIDE;
        const float *Wbeta = P(base + 0);
        const float *We = P(base + 1),  *be  = P(base + 2);
        const float *Wf1 = P(base + 3), *bf1 = P(base + 4);
        const float *Wf2 = P(base + 5), *bf2 = P(base + 6);
        const float *Wk = P(base + 7),  *bk  = P(base + 8);
        const float *ln1b = P(base + 9),  *ln1g = P(base + 10);
        const float *ln2b = P(base + 11), *ln2g = P(base + 12);
        const float *Wq = P(base + 13), *bq = P(base + 14);
        const float *Wsk = P(base + 15), *bsk = P(base + 16);
        const float *Wv = P(base + 17), *bv = P(base + 18);

        layernorm_kernel<<<gWave(N_), TPB, 0, stream>>>(H, HN, ln1g, ln1b, N_);

        gemm_lds<<<gGemm(N_, 256), TPB, 0, stream>>>(HN, Wq,  bq,  Qb, N_, 256, 256, 0);
        gemm_lds<<<gGemm(N_, 256), TPB, 0, stream>>>(HN, Wk,  bk,  Kb, N_, 256, 256, 0);
        gemm_lds<<<gGemm(N_, 256), TPB, 0, stream>>>(HN, Wv,  bv,  Vb, N_, 256, 256, 0);
        gemm_lds<<<gGemm(N_, 256), TPB, 0, stream>>>(HN, Wsk, bsk, XR, N_, 256, 256, 0);

        fill_u32<<<gElem((long)N_ * 256), TPB, 0, stream>>>((unsigned*)ATT, 0u, (long)N_ * 256);
        fill_u32<<<gElem((long)N_ * 8), TPB, 0, stream>>>(MB, 0u, (long)N_ * 8);
        fill_u32<<<gElem((long)N_ * 8), TPB, 0, stream>>>((unsigned*)SB, 0u, (long)N_ * 8);

        attn_score<<<gWave(E_), TPB, 0, stream>>>(Qb, Kb, eattr, We, be, src, dst, ALPHA, MB, E_);
        attn_expsum<<<gElem((long)E_ * 8), TPB, 0, stream>>>(dst, ALPHA, MB, SB, E_);
        attn_aggregate<<<gWave(E_), TPB, 0, stream>>>(Vb, eattr, We, be, src, dst, ALPHA, SB, ATT, E_);
        beta_combine<<<gWave(N_), TPB, 0, stream>>>(ATT, XR, Wbeta, H, N_);

        layernorm_kernel<<<gWave(N_), TPB, 0, stream>>>(H, HN, ln2g, ln2b, N_);
        gemm_lds<<<gGemm(N_, 1024), TPB, 0, stream>>>(HN, Wf1, bf1, FFH, N_, 256, 1024, GEMM_RELU);
        gemm_lds<<<gGemm(N_, 256), TPB, 0, stream>>>(FFH, Wf2, bf2, H, N_, 1024, 256, GEMM_ACC);

        copy_cat<<<gElem((long)N_ * 256), TPB, 0, stream>>>(H, XCAT, N_, b);
    }

    // gating + pooling
    gemm_lds<<<gGemm(N_, 256), TPB, 0, stream>>>(XCAT, P(44), P(45), ATT, N_, 512, 256, GEMM_RELU);
    fill_u32<<<1, TPB, 0, stream>>>(GMAX, 0u, NGRAPH);
    fill_u32<<<1, TPB, 0, stream>>>((unsigned*)GSUM, 0u, NGRAPH);
    fill_u32<<<gElem((long)NGRAPH * 512), TPB, 0, stream>>>((unsigned*)POOL, 0u, (long)NGRAPH * 512);
    gate_score<<<gWave(N_), TPB, 0, stream>>>(ATT, P(46), P(47), batch, GSC, GMAX, N_);
    gate_expsum<<<gElem(N_), TPB, 0, stream>>>(batch, GSC, GMAX, GSUM, N_);
    pool_scatter<<<gWave(N_), TPB, 0, stream>>>(XCAT, GSC, GSUM, batch, POOL, N_);

    gemm_lds<<<gGemm(NGRAPH, 256), TPB, 0, stream>>>(POOL, P(48), P(49), HH, NGRAPH, 512, 256, GEMM_RELU);
    head2_kernel<<<1, 256, 0, stream>>>(HH, P(50), P(51), (float*)d_out);
}